// GATLayer_7009386627237
// MI455X (gfx1250) — compile-verified
//
#include <hip/hip_runtime.h>

#define IN_F 64
#define OUT_F 64
#define EDGE_DIM 16
#define NEG_SLOPE 0.2f

typedef __attribute__((ext_vector_type(2))) float v2f;
typedef __attribute__((ext_vector_type(8))) float v8f;

// Monotonic int encoding of f32 so signed-int atomicMax == float max.
__device__ __forceinline__ int order_f32(float f) {
  int i = __float_as_int(f);
  return (i >= 0) ? i : (i ^ 0x7fffffff);
}
__device__ __forceinline__ float unorder_f32(int i) {
  return __int_as_float((i >= 0) ? i : (i ^ 0x7fffffff));
}

// ---------------------------------------------------------------- init
__global__ void k_init(float* __restrict__ out, int out_n,
                       float* __restrict__ denom, int* __restrict__ m_i, int n_nodes) {
  int gid = blockIdx.x * blockDim.x + threadIdx.x;
  if (gid < out_n) out[gid] = 0.0f;
  if (gid < n_nodes) {
    denom[gid] = 0.0f;
    m_i[gid]   = (int)0x80000000;  // encodes -inf ordering floor
  }
}

// ---------------------------------------------------------------- xt = x @ W
// One wave computes a 16x64 stripe of xt via V_WMMA_F32_16X16X4_F32.
// A (16x4 f32): lanes 0-15 -> M, a.x = K, a.y = K+1; lanes 16-31 use K+2,K+3.
// C/D (16x16 f32): VGPR r -> row (r + 8*hi), col (lane & 15).
__global__ void k_gemm_wmma(const float* __restrict__ x, const float* __restrict__ W,
                            float* __restrict__ xt, int n_nodes) {
  const int wave = (blockIdx.x * blockDim.x + threadIdx.x) >> 5;
  const int lane = threadIdx.x & 31;
  const int row0 = wave * 16;
  if (row0 >= n_nodes) return;
  const int hi  = lane >> 4;
  const int l15 = lane & 15;
  const int arow = row0 + l15;

  v8f acc0 = {}; v8f acc1 = {}; v8f acc2 = {}; v8f acc3 = {};
#pragma unroll
  for (int kc = 0; kc < IN_F; kc += 4) {
    const int k = kc + (hi ? 2 : 0);
    v2f a;
    a.x = x[arow * IN_F + k];
    a.y = x[arow * IN_F + k + 1];
    const float* wr0 = W + k * OUT_F + l15;
    const float* wr1 = W + (k + 1) * OUT_F + l15;
    v2f b0; b0.x = wr0[0];  b0.y = wr1[0];
    v2f b1; b1.x = wr0[16]; b1.y = wr1[16];
    v2f b2; b2.x = wr0[32]; b2.y = wr1[32];
    v2f b3; b3.x = wr0[48]; b3.y = wr1[48];
    acc0 = __builtin_amdgcn_wmma_f32_16x16x4_f32(false, a, false, b0, (short)0, acc0, false, false);
    acc1 = __builtin_amdgcn_wmma_f32_16x16x4_f32(false, a, false, b1, (short)0, acc1, false, false);
    acc2 = __builtin_amdgcn_wmma_f32_16x16x4_f32(false, a, false, b2, (short)0, acc2, false, false);
    acc3 = __builtin_amdgcn_wmma_f32_16x16x4_f32(false, a, false, b3, (short)0, acc3, false, false);
  }
#pragma unroll
  for (int r = 0; r < 8; ++r) {
    const int row = row0 + r + 8 * hi;
    float* orow = xt + row * OUT_F + l15;
    orow[0]  = acc0[r];
    orow[16] = acc1[r];
    orow[32] = acc2[r];
    orow[48] = acc3[r];
  }
}

// ---------------------------------------------------------------- per-node dots
// One wave per node: s = <xt[i], a_src>, d = <xt[i], a_dst>.
__global__ void k_node_dots(const float* __restrict__ xt, const float* __restrict__ a,
                            float* __restrict__ s_node, float* __restrict__ d_node,
                            int n_nodes) {
  const int wave = (blockIdx.x * blockDim.x + threadIdx.x) >> 5;
  const int lane = threadIdx.x & 31;
  if (wave >= n_nodes) return;
  const float* row = xt + (long)wave * OUT_F;
  float v0 = row[lane], v1 = row[lane + 32];
  float s = v0 * a[lane]         + v1 * a[lane + 32];
  float d = v0 * a[OUT_F + lane] + v1 * a[OUT_F + lane + 32];
#pragma unroll
  for (int off = 16; off > 0; off >>= 1) {
    s += __shfl_down(s, off, 32);
    d += __shfl_down(d, off, 32);
  }
  if (lane == 0) { s_node[wave] = s; d_node[wave] = d; }
}

// ---------------------------------------------------------------- per-edge contrib
__global__ void k_edge_contrib(const float* __restrict__ edge_attr, const float* __restrict__ a,
                               float* __restrict__ e_c, int n_edges) {
  const int e = blockIdx.x * blockDim.x + threadIdx.x;
  if (e >= n_edges) return;
  const float4* row = (const float4*)(edge_attr + (long)e * EDGE_DIM);
  const float* ae = a + 2 * OUT_F;
  float acc = 0.0f;
#pragma unroll
  for (int q = 0; q < 4; ++q) {
    float4 v = row[q];
    acc += v.x * ae[4*q] + v.y * ae[4*q+1] + v.z * ae[4*q+2] + v.w * ae[4*q+3];
  }
  e_c[e] = acc;
}

// Augmented edge list: [0,E) fwd, [E,2E) reversed, [2E,2E+N) self-loops.
__device__ __forceinline__ void aug_edge(int gid, const int* __restrict__ ei,
                                         int n_edges, int& s_idx, int& d_idx, int& raw) {
  if (gid < n_edges)            { raw = gid;               s_idx = ei[raw];           d_idx = ei[n_edges + raw]; }
  else if (gid < 2 * n_edges)   { raw = gid - n_edges;     s_idx = ei[n_edges + raw]; d_idx = ei[raw]; }
  else                          { raw = -1; s_idx = d_idx = gid - 2 * n_edges; }
}

// ---------------------------------------------------------------- logits + segment max
__global__ void k_logits_max(const int* __restrict__ ei, const float* __restrict__ e_c,
                             const float* __restrict__ s_node, const float* __restrict__ d_node,
                             float* __restrict__ zbuf, int* __restrict__ m_i,
                             int n_edges, int n_nodes) {
  const int gid = blockIdx.x * blockDim.x + threadIdx.x;
  if (gid >= 2 * n_edges + n_nodes) return;
  int s_idx, d_idx, raw;
  aug_edge(gid, ei, n_edges, s_idx, d_idx, raw);
  const float e = (raw >= 0) ? e_c[raw] : 0.0f;
  const float v = s_node[s_idx] + d_node[d_idx] + e;
  const float logit = (v >= 0.0f) ? v : NEG_SLOPE * v;   // T == 1.0
  zbuf[gid] = logit;
  atomicMax(&m_i[d_idx], order_f32(logit));
}

// ---------------------------------------------------------------- z = exp(logit - m), denom
__global__ void k_exp_sum(const int* __restrict__ ei, const int* __restrict__ m_i,
                          float* __restrict__ zbuf, float* __restrict__ denom,
                          int n_edges, int n_nodes) {
  const int gid = blockIdx.x * blockDim.x + threadIdx.x;
  if (gid >= 2 * n_edges + n_nodes) return;
  int s_idx, d_idx, raw;
  aug_edge(gid, ei, n_edges, s_idx, d_idx, raw);
  const float m = unorder_f32(m_i[d_idx]);
  const float z = __expf(zbuf[gid] - m);
  zbuf[gid] = z;
  atomicAdd(&denom[d_idx], z);
}

// ---------------------------------------------------------------- weighted scatter-add
// One thread per (edge, feature): coalesced gather of xt[src], coalesced f32 atomics to out[dst].
__global__ void k_scatter(const int* __restrict__ ei, const float* __restrict__ xt,
                          const float* __restrict__ zbuf, const float* __restrict__ denom,
                          float* __restrict__ out, int n_edges, int n_nodes) {
  const long gid = (long)blockIdx.x * blockDim.x + threadIdx.x;
  const long total = (long)(2 * n_edges + n_nodes) * OUT_F;
  if (gid >= total) return;
  const int e = (int)(gid >> 6);
  const int j = (int)(gid & 63);
  int s_idx, d_idx, raw;
  aug_edge(e, ei, n_edges, s_idx, d_idx, raw);
  const float att = zbuf[e] / denom[d_idx];
  atomicAdd(&out[(long)d_idx * OUT_F + j], att * xt[(long)s_idx * OUT_F + j]);
}

// ---------------------------------------------------------------- launcher
extern "C" void kernel_launch(void* const* d_in, const int* in_sizes, int n_in,
                              void* d_out, int out_size, void* d_ws, size_t ws_size,
                              hipStream_t stream) {
  const float* x         = (const float*)d_in[0];
  const int*   ei        = (const int*)d_in[1];
  const float* edge_attr = (const float*)d_in[2];
  // d_in[3] = batch_id (unused by the reference output)
  const float* W         = (const float*)d_in[4];
  const float* a         = (const float*)d_in[5];

  const int N  = in_sizes[0] / IN_F;     // 50000
  const int E  = in_sizes[1] / 2;        // 800000
  const int EA = 2 * E + N;              // augmented edge count

  // Workspace layout (floats): xt | s | d | e_c | m_i | denom | zbuf
  float* xt     = (float*)d_ws;
  float* s_node = xt + (long)N * OUT_F;
  float* d_node = s_node + N;
  float* e_c    = d_node + N;
  int*   m_i    = (int*)(e_c + E);
  float* denom  = (float*)(m_i + N);
  float* zbuf   = denom + N;
  float* out    = (float*)d_out;

  const int B = 256;

  // 0) init accumulators (out is accumulated into -> must zero every launch)
  {
    int n = (out_size > N) ? out_size : N;
    k_init<<<(n + B - 1) / B, B, 0, stream>>>(out, out_size, denom, m_i, N);
  }
  // 1) xt = x @ W  (WMMA f32 16x16x4, one wave per 16-row tile)
  {
    int waves = (N + 15) / 16;
    int thr = waves * 32;
    k_gemm_wmma<<<(thr + B - 1) / B, B, 0, stream>>>(x, W, xt, N);
  }
  // 2) per-node attention dots (wave per node)
  {
    long thr = (long)N * 32;
    k_node_dots<<<(int)((thr + B - 1) / B), B, 0, stream>>>(xt, a, s_node, d_node, N);
  }
  // 3) per-edge attr contribution
  k_edge_contrib<<<(E + B - 1) / B, B, 0, stream>>>(edge_attr, a, e_c, E);
  // 4) logits + segment max (int-ordered atomicMax)
  k_logits_max<<<(EA + B - 1) / B, B, 0, stream>>>(ei, e_c, s_node, d_node, zbuf, m_i, E, N);
  // 5) exp + segment sum
  k_exp_sum<<<(EA + B - 1) / B, B, 0, stream>>>(ei, m_i, zbuf, denom, E, N);
  // 6) att-weighted feature scatter
  {
    long thr = (long)EA * OUT_F;
    k_scatter<<<(int)((thr + B - 1) / B), B, 0, stream>>>(ei, xt, zbuf, denom, out, E, N);
  }
}